// BERT_CRF_NER_44135083933994
// MI455X (gfx1250) — compile-verified
//
#include <hip/hip_runtime.h>
#include <hip/hip_bf16.h>

// ---------------------------------------------------------------------------
// BERT-CRF NER on MI455X (gfx1250):
//   Kernel 1: feats[b,t,l] = hidden[b,t,:] @ W[:,l] + bias[l]
//             done as 16x16 output tiles with V_WMMA_F32_16X16X4_F32,
//             labels padded 9 -> 16. Memory-bound on the 96MB hidden read.
//   Kernel 2: per-batch Viterbi scan (wave32 per batch, lane = "to" label),
//             psi stored as bytes, backtrack on lane 0, path out as float.
// ---------------------------------------------------------------------------

typedef __attribute__((ext_vector_type(2))) float v2f;
typedef __attribute__((ext_vector_type(8))) float v8f;

#define HID   768
#define NLAB  9
#define LPAD  16
#define TLEN  512
#define BATCH 64
#define NEGINF -10000.0f
#define START_LABEL 7

// ---------------- Kernel 1: WMMA GEMM  [32768 x 768] x [768 x 16] ----------
__global__ __launch_bounds__(256) void feats_gemm_wmma(
    const float* __restrict__ hs,    // [B*T, 768]
    const float* __restrict__ W,     // [768, 9] row-major
    const float* __restrict__ bias,  // [9]
    float* __restrict__ feats)       // [B*T, 16] padded
{
    // Stage W zero-padded to 16 columns in LDS: 768*16*4 = 48KB.
    __shared__ float Wpad[HID * LPAD];
    const int tid = threadIdx.x;
    for (int i = tid; i < HID * LPAD; i += 256) {
        int k = i >> 4;
        int n = i & 15;
        Wpad[i] = (n < NLAB) ? W[k * NLAB + n] : 0.0f;
    }
    __syncthreads();

    const int wave  = tid >> 5;              // 8 waves per block
    const int lane  = tid & 31;
    const int tile  = blockIdx.x * 8 + wave; // 2048 tiles of 16 rows
    const int m     = lane & 15;             // A row / B,C column index
    const int khalf = lane >> 4;             // 0: K=k0,k0+1  1: K=k0+2,k0+3

    const long row = (long)tile * 16 + m;
    const float* __restrict__ arow = hs + row * HID + 2 * khalf;
    const float* __restrict__ wcol = &Wpad[2 * khalf * LPAD + m];

    v8f c = {0.f, 0.f, 0.f, 0.f, 0.f, 0.f, 0.f, 0.f};

    #pragma unroll 4
    for (int k0 = 0; k0 < HID; k0 += 4) {
        // A 16x4 fp32: contiguous pair per lane (global_load_b64)
        v2f a = *(const v2f*)(arow + k0);
        // B 4x16 fp32 from LDS (zero-padded cols keep EXEC uniform)
        v2f b;
        b.x = wcol[(k0 + 0) * LPAD];
        b.y = wcol[(k0 + 1) * LPAD];
        // D = A*B + C  (v_wmma_f32_16x16x4_f32)
        c = __builtin_amdgcn_wmma_f32_16x16x4_f32(
                /*neg_a=*/false, a, /*neg_b=*/false, b,
                /*c_mod=*/(short)0, c, /*reuse_a=*/false, /*reuse_b=*/false);
    }

    // C/D layout: VGPR r -> row r (lanes 0-15) / row r+8 (lanes 16-31), col = m
    const float bv = (m < NLAB) ? bias[m] : 0.0f;
    const long base = (long)tile * 16;
    #pragma unroll
    for (int r = 0; r < 8; ++r) {
        int outrow = r + khalf * 8;
        feats[(base + outrow) * LPAD + m] = c[r] + bv;
    }
}

// ---------------- Kernel 2: Viterbi scan + backtrack, one wave per batch ----
__global__ __launch_bounds__(32) void viterbi_decode(
    const float* __restrict__ feats,  // [B*T, 16] padded
    const float* __restrict__ trans,  // [9, 9]  trans[to, from]
    unsigned char* __restrict__ psi,  // [T-1, B, 9]
    float* __restrict__ out)          // [B, T] path as float
{
    const int b    = blockIdx.x;   // 64 batches
    const int lane = threadIdx.x;  // 32 lanes, lanes 0..8 = "to" label
    const int to   = lane;

    // cache transition row into registers
    float trow[NLAB];
    #pragma unroll
    for (int f = 0; f < NLAB; ++f)
        trow[f] = (to < NLAB) ? trans[to * NLAB + f] : NEGINF;

    float ld = (to == START_LABEL) ? 0.0f : NEGINF;

    for (int t = 1; t < TLEN; ++t) {
        float best = -3.0e38f;
        int   arg  = 0;
        #pragma unroll
        for (int f = 0; f < NLAB; ++f) {
            float ldf = __shfl(ld, f, 32);   // broadcast log_delta[from=f]
            float s   = trow[f] + ldf;
            if (s > best) { best = s; arg = f; }  // strict > == argmax-first
        }
        if (to < NLAB) {
            ld = best + feats[((long)b * TLEN + t) * LPAD + to];
            psi[((long)(t - 1) * BATCH + b) * NLAB + to] = (unsigned char)arg;
        }
    }

    // argmax over final log_delta (same result on all lanes)
    float bestf = -3.0e38f;
    int   last  = 0;
    #pragma unroll
    for (int f = 0; f < NLAB; ++f) {
        float v = __shfl(ld, f, 32);
        if (v > bestf) { bestf = v; last = f; }
    }

    if (lane == 0) {
        out[(long)b * TLEN + (TLEN - 1)] = (float)last;
        int p = last;
        for (int t = TLEN - 2; t >= 0; --t) {
            p = (int)psi[((long)t * BATCH + b) * NLAB + p];
            out[(long)b * TLEN + t] = (float)p;
        }
    }
}

// ---------------------------------------------------------------------------
extern "C" void kernel_launch(void* const* d_in, const int* in_sizes, int n_in,
                              void* d_out, int out_size, void* d_ws, size_t ws_size,
                              hipStream_t stream) {
    const float* hs    = (const float*)d_in[0];  // [64,512,768]
    const float* W     = (const float*)d_in[1];  // [768,9]
    const float* bias  = (const float*)d_in[2];  // [9]
    const float* trans = (const float*)d_in[3];  // [9,9]
    float* out = (float*)d_out;                  // [64,512]

    // workspace: padded feats then psi bytes
    float* feats = (float*)d_ws;                                     // 32768*16 f32 = 2MB
    unsigned char* psi = (unsigned char*)d_ws + (size_t)32768 * LPAD * sizeof(float);

    // 2048 row-tiles of 16, 8 waves per 256-thread block -> 256 blocks
    feats_gemm_wmma<<<256, 256, 0, stream>>>(hs, W, bias, feats);
    viterbi_decode<<<BATCH, 32, 0, stream>>>(feats, trans, psi, out);
}